// ImageWarped_76854144795315
// MI455X (gfx1250) — compile-verified
//
#include <hip/hip_runtime.h>

// Trilinear 3D grid-sample for MI455X (gfx1250).
// image: [B=2, 128,128,128, C=1] f32 (16 MB -> resident in 192MB L2)
// grid:  [B=2, N=2^21, 3] f32 (48 MB, streamed once, NT)
// out:   [B=2, N=2^21, 1] f32 (16 MB, streamed once, NT)
// Memory-bound: ~80 MB HBM floor ~= 3.4us @ 23.3 TB/s. No matmul structure,
// so no WMMA; the CDNA5 wins are NT temporal hints (protect image L2
// residency), global_prefetch_b8 on the grid stream, and wave32-coalesced
// b96 grid loads.

#define VOL   128
#define VOL3  (VOL * VOL * VOL)       // 2,097,152
#define NPTS  2097152u                // N per batch
#define NPTS_SHIFT 21                 // NPTS == 1<<21
#define TOTAL (2u * NPTS)             // 4,194,304 samples

__global__ __launch_bounds__(256) void trilerp3d_kernel(
    const float* __restrict__ image,
    const float* __restrict__ grid,
    float* __restrict__ out)
{
    const unsigned stride = blockDim.x * gridDim.x;
    unsigned i = blockIdx.x * blockDim.x + threadIdx.x;

    for (; i < TOTAL; i += stride) {
        // Prefetch next iteration's grid coords into cache (speculative,
        // streaming read) -> global_prefetch_b8. Distance = one grid-stride
        // iteration, which spans 8 dependent L2-gather round trips of work.
        unsigned nxt = i + stride;
        if (nxt < TOTAL) {
            __builtin_prefetch(&grid[(size_t)nxt * 3u], 0, 0);
        }

        // --- streamed grid read (non-temporal: do not pollute L2) ---
        const float* g3 = grid + (size_t)i * 3u;
        float gx = __builtin_nontemporal_load(g3 + 0);
        float gy = __builtin_nontemporal_load(g3 + 1);
        float gz = __builtin_nontemporal_load(g3 + 2);

        // clip(grid * 128, 0.001, 126.999) exactly as reference
        float x = fminf(fmaxf(gx * 128.0f, 0.001f), 126.999f);
        float y = fminf(fmaxf(gy * 128.0f, 0.001f), 126.999f);
        float z = fminf(fmaxf(gz * 128.0f, 0.001f), 126.999f);

        float x1f = floorf(x), x2f = ceilf(x);
        float y1f = floorf(y), y2f = ceilf(y);
        float z1f = floorf(z), z2f = ceilf(z);

        int ix1 = (int)x1f, ix2 = (int)x2f;
        int iy1 = (int)y1f, iy2 = (int)y2f;
        int iz1 = (int)z1f, iz2 = (int)z2f;

        // batch index and image base (C == 1)
        unsigned b = i >> NPTS_SHIFT;
        const float* base = image + (size_t)b * (size_t)VOL3;

        // row offsets: x stride = 128*128, y stride = 128, z stride = 1
        int ox1 = ix1 << 14, ox2 = ix2 << 14;   // * 16384
        int oy1 = iy1 << 7,  oy2 = iy2 << 7;    // * 128

        // 8 independent gathers (image resident in L2; default RT temporal
        // hint keeps these lines cached while NT streams pass through)
        float g111 = base[ox1 + oy1 + iz1];
        float g211 = base[ox2 + oy1 + iz1];
        float g121 = base[ox1 + oy2 + iz1];
        float g221 = base[ox2 + oy2 + iz1];
        float g112 = base[ox1 + oy1 + iz2];
        float g212 = base[ox2 + oy1 + iz2];
        float g122 = base[ox1 + oy2 + iz2];
        float g222 = base[ox2 + oy2 + iz2];

        // weights exactly as reference (note: ceil==floor => both weights 0)
        float wx = x - x1f, wx2 = x2f - x;
        float wy = y - y1f, wy2 = y2f - y;
        float wz = z - z1f, wz2 = z2f - z;

        float ly1 = (g211 * wx + g111 * wx2) * wy2
                  + (g221 * wx + g121 * wx2) * wy;
        float ly2 = (g212 * wx + g112 * wx2) * wy2
                  + (g222 * wx + g122 * wx2) * wy;
        float res = ly2 * wz + ly1 * wz2;

        // streamed output write (non-temporal)
        __builtin_nontemporal_store(res, out + i);
    }
}

extern "C" void kernel_launch(void* const* d_in, const int* in_sizes, int n_in,
                              void* d_out, int out_size, void* d_ws, size_t ws_size,
                              hipStream_t stream) {
    (void)in_sizes; (void)n_in; (void)d_ws; (void)ws_size; (void)out_size;
    const float* image = (const float*)d_in[0];
    const float* grid  = (const float*)d_in[1];
    float* out = (float*)d_out;

    // 2048 blocks x 256 threads (8 waves/block, wave32) = 524,288 threads,
    // 8 grid-stride iterations each -> deep load-latency hiding, and the
    // prefetch covers 7 of 8 iterations.
    dim3 block(256);
    dim3 gridDim(2048);
    trilerp3d_kernel<<<gridDim, block, 0, stream>>>(image, grid, out);
}